// AttentionCondenser_63797444215338
// MI455X (gfx1250) — compile-verified
//
#include <hip/hip_runtime.h>
#include <stdint.h>
#include <math.h>

typedef __attribute__((ext_vector_type(2))) float        v2f;
typedef __attribute__((ext_vector_type(8))) float        v8f;
typedef __attribute__((ext_vector_type(4))) unsigned int v4u;
typedef __attribute__((ext_vector_type(8))) int          v8i;
typedef __attribute__((ext_vector_type(4))) int          v4i;

#define CCH   512   // channels
#define NPIX  1024  // h*w
#define BATCH 32

// ---------------------------------------------------------------------------
// TDM: DMA a (tile_h x tile_w) fp32 tile (row pitch = stride elements) from
// global memory into LDS at byte offset lds_off. 2-D descriptor per
// CDNA5 ISA. Issued by one wave; completion
// tracked with TENSORcnt. 6-arg builtin form (clang-23 / therock headers).
// ---------------------------------------------------------------------------
__device__ __forceinline__ void tdm_load_tile(unsigned lds_off,
                                              const float* gptr,
                                              unsigned tensor_w, unsigned tensor_h,
                                              unsigned tile_w, unsigned tile_h,
                                              unsigned stride_elems) {
  unsigned long long ga = (unsigned long long)(uintptr_t)gptr;
  v4u g0 = {
      1u,                                            // count=1 (valid), user D#
      lds_off,                                       // lds_addr (bytes)
      (unsigned)(ga & 0xFFFFFFFFu),                  // global_addr[31:0]
      (unsigned)((ga >> 32) & 0x01FFFFFFu) | (2u << 30)  // addr[56:32], type=2
  };
  v8i g1;
  g1[0] = 0x20000;                                   // data_size=2 -> 4 bytes
  g1[1] = (int)((tensor_w & 0xFFFFu) << 16);         // tensor_dim0[15:0] @ bits 63:48
  g1[2] = (int)(((tensor_w >> 16) & 0xFFFFu) |       // tensor_dim0[31:16]
                ((tensor_h & 0xFFFFu) << 16));       // tensor_dim1[15:0]
  g1[3] = (int)(((tensor_h >> 16) & 0xFFFFu) |       // tensor_dim1[31:16]
                ((tile_w & 0xFFFFu) << 16));         // tile_dim0
  g1[4] = (int)(tile_h & 0xFFFFu);                   // tile_dim1 (tile_dim2=0)
  g1[5] = (int)stride_elems;                         // tensor_dim0_stride[31:0]
  g1[6] = 0;                                         // stride hi / dim1_stride
  g1[7] = 0;
  v4i gz4 = {0, 0, 0, 0};
  v8i gz8 = {0, 0, 0, 0, 0, 0, 0, 0};
  __builtin_amdgcn_tensor_load_to_lds(g0, g1, gz4, gz4, gz8, 0);
}

// ---------------------------------------------------------------------------
// Stage 1: y[b,o,n] = sum_c W[o,c]*x[b,c,n] + bias[o]
// Block: 8 waves -> 128(o) x 128(n) tile; wave: 32x64.
// x K-slab (16 x 128) staged to LDS by TDM, double buffered.
// W fragments are contiguous float2 global loads.
// ---------------------------------------------------------------------------
__global__ void __launch_bounds__(256) k_gemm_y(const float* __restrict__ X,
                                                const float* __restrict__ W,
                                                const float* __restrict__ bias,
                                                float* __restrict__ Y) {
  extern __shared__ float lds[];                     // 2 x (16*128) floats
  const int lane = threadIdx.x & 31;
  const int hi   = (lane >> 4) & 1;
  const int l15  = lane & 15;
  const int w    = threadIdx.x >> 5;                 // wave in block
  const int wm   = w >> 1, wn = w & 1;

  const int blk = blockIdx.x;                        // 32 blocks per batch
  const int b   = blk >> 5;
  const int rem = blk & 31;
  const int m0B = (rem >> 3) << 7;                   // 4 m-tiles of 128
  const int n0B = (rem & 7) << 7;                    // 8 n-tiles of 128

  const int m0w = m0B + wm * 32;
  const int n0w = wn * 64;                           // local within block tile

  const float* x = X + (size_t)b * CCH * NPIX;
  float*       y = Y + (size_t)b * CCH * NPIX;

  v8f acc[2][4] = {};
  const int arow0 = m0w + l15;
  const int arow1 = arow0 + 16;

  // prologue: DMA K-chunk 0
  if (threadIdx.x < 32) {
    tdm_load_tile(0u, x + n0B, NPIX, CCH, 128, 16, NPIX);
    __builtin_amdgcn_s_wait_tensorcnt(0);
  }
  __syncthreads();

  for (int kc = 0; kc < CCH / 16; ++kc) {
    const float* lbuf = lds + (kc & 1) * 2048;
    // prefetch next K-chunk into the other buffer
    if ((kc + 1) < CCH / 16 && threadIdx.x < 32) {
      tdm_load_tile((unsigned)(((kc + 1) & 1) * 8192),
                    x + (size_t)(kc + 1) * 16 * NPIX + n0B,
                    NPIX, CCH, 128, 16, NPIX);
    }
#pragma unroll
    for (int k4 = 0; k4 < 4; ++k4) {
      const int kg = kc * 16 + k4 * 4 + 2 * hi;      // global K of this lane pair
      const int kl = k4 * 4 + 2 * hi;                // K within LDS slab
      v2f a0 = *(const v2f*)(W + (size_t)arow0 * CCH + kg);
      v2f a1 = *(const v2f*)(W + (size_t)arow1 * CCH + kg);
#pragma unroll
      for (int t = 0; t < 4; ++t) {
        const int col = n0w + t * 16 + l15;
        v2f bf;
        bf.x = lbuf[kl * 128 + col];
        bf.y = lbuf[(kl + 1) * 128 + col];
        acc[0][t] = __builtin_amdgcn_wmma_f32_16x16x4_f32(
            false, a0, false, bf, (short)0, acc[0][t], false, false);
        acc[1][t] = __builtin_amdgcn_wmma_f32_16x16x4_f32(
            false, a1, false, bf, (short)0, acc[1][t], false, false);
      }
    }
    if (threadIdx.x < 32) __builtin_amdgcn_s_wait_tensorcnt(0);
    __syncthreads();
  }

  float bsv[2][8];
#pragma unroll
  for (int mi = 0; mi < 2; ++mi)
#pragma unroll
    for (int v = 0; v < 8; ++v) bsv[mi][v] = bias[m0w + mi * 16 + v + 8 * hi];

#pragma unroll
  for (int mi = 0; mi < 2; ++mi)
#pragma unroll
    for (int t = 0; t < 4; ++t)
#pragma unroll
      for (int v = 0; v < 8; ++v) {
        const int row = m0w + mi * 16 + v + 8 * hi;
        const int col = n0B + n0w + t * 16 + l15;
        y[(size_t)row * NPIX + col] = acc[mi][t][v] + bsv[mi][v];
      }
}

// ---------------------------------------------------------------------------
// Stage 2: S[b,i,j] = sum_n y[b,i,n]*y[b,j,n]   (y @ y^T, K = NPIX)
// Wave: 32x64; all fragments are contiguous float2 global loads (L2-resident).
// ---------------------------------------------------------------------------
__global__ void __launch_bounds__(256) k_gemm_s(const float* __restrict__ Y,
                                                float* __restrict__ S) {
  const int lane = threadIdx.x & 31;
  const int hi   = (lane >> 4) & 1;
  const int l15  = lane & 15;
  const int wid  = blockIdx.x * (blockDim.x >> 5) + (threadIdx.x >> 5);

  const int b   = wid >> 7;            // 128 waves per batch
  const int rem = wid & 127;
  const int m0  = (rem >> 3) << 5;     // 16 m-tiles of 32
  const int n0  = (rem & 7) << 6;      // 8 n-strips of 64

  const float* y = Y + (size_t)b * CCH * NPIX;
  float*       s = S + (size_t)b * CCH * CCH;

  v8f acc[2][4] = {};
  const int arow0 = m0 + l15;
  const int arow1 = arow0 + 16;

  for (int k = 0; k < NPIX; k += 4) {
    const int kk = k + 2 * hi;
    v2f a0 = *(const v2f*)(y + (size_t)arow0 * NPIX + kk);
    v2f a1 = *(const v2f*)(y + (size_t)arow1 * NPIX + kk);
#pragma unroll
    for (int t = 0; t < 4; ++t) {
      const int col = n0 + t * 16 + l15;
      v2f bf = *(const v2f*)(y + (size_t)col * NPIX + kk);
      acc[0][t] = __builtin_amdgcn_wmma_f32_16x16x4_f32(
          false, a0, false, bf, (short)0, acc[0][t], false, false);
      acc[1][t] = __builtin_amdgcn_wmma_f32_16x16x4_f32(
          false, a1, false, bf, (short)0, acc[1][t], false, false);
    }
  }

#pragma unroll
  for (int mi = 0; mi < 2; ++mi)
#pragma unroll
    for (int t = 0; t < 4; ++t)
#pragma unroll
      for (int v = 0; v < 8; ++v) {
        const int row = m0 + mi * 16 + v + 8 * hi;
        const int col = n0 + t * 16 + l15;
        s[(size_t)row * CCH + col] = acc[mi][t][v];
      }
}

// ---------------------------------------------------------------------------
// Stage 3: row softmax over S (rows of 512), one wave32 per row, in-place.
// ---------------------------------------------------------------------------
__global__ void __launch_bounds__(256) k_softmax(float* __restrict__ S) {
  const int lane = threadIdx.x & 31;
  const int wid  = blockIdx.x * (blockDim.x >> 5) + (threadIdx.x >> 5);
  float* row = S + (size_t)wid * CCH;   // wid in [0, BATCH*CCH)

  float vals[16];
  float m = -3.4e38f;
#pragma unroll
  for (int i = 0; i < 16; ++i) {
    vals[i] = row[lane + i * 32];
    m = fmaxf(m, vals[i]);
  }
#pragma unroll
  for (int off = 16; off > 0; off >>= 1)
    m = fmaxf(m, __shfl_xor(m, off, 32));

  float sum = 0.0f;
#pragma unroll
  for (int i = 0; i < 16; ++i) {
    vals[i] = __expf(vals[i] - m);
    sum += vals[i];
  }
#pragma unroll
  for (int off = 16; off > 0; off >>= 1)
    sum += __shfl_xor(sum, off, 32);

  const float inv = 1.0f / sum;
#pragma unroll
  for (int i = 0; i < 16; ++i)
    row[lane + i * 32] = vals[i] * inv;
}

// ---------------------------------------------------------------------------
// Stage 4: out[b,n,d] = sum_c y[b,c,n] * A[b,c,d]   (y^T @ A, K = CCH)
// Block: 8 waves -> 128(n) x 128(d); wave: 32x64.
// A(tt) K-slab (16 x 128) staged to LDS by TDM, double buffered.
// y^T fragments are strided scalar loads (L2-resident).
// ---------------------------------------------------------------------------
__global__ void __launch_bounds__(256) k_gemm_out(const float* __restrict__ Y,
                                                  const float* __restrict__ Att,
                                                  float* __restrict__ Out) {
  extern __shared__ float lds[];
  const int lane = threadIdx.x & 31;
  const int hi   = (lane >> 4) & 1;
  const int l15  = lane & 15;
  const int w    = threadIdx.x >> 5;
  const int wm   = w >> 1, wn = w & 1;

  const int blk = blockIdx.x;                        // 32 blocks per batch
  const int b   = blk >> 5;
  const int rem = blk & 31;
  const int m0B = (rem >> 2) << 7;                   // 8 m-tiles of 128 (n dim)
  const int n0B = (rem & 3) << 7;                    // 4 n-tiles of 128 (d dim)

  const int m0w = m0B + wm * 32;
  const int n0w = wn * 64;

  const float* y   = Y   + (size_t)b * CCH * NPIX;
  const float* att = Att + (size_t)b * CCH * CCH;
  float*       out = Out + (size_t)b * NPIX * CCH;

  v8f acc[2][4] = {};
  const int mrow0 = m0w + l15;
  const int mrow1 = mrow0 + 16;

  if (threadIdx.x < 32) {
    tdm_load_tile(0u, att + n0B, CCH, CCH, 128, 16, CCH);
    __builtin_amdgcn_s_wait_tensorcnt(0);
  }
  __syncthreads();

  for (int kc = 0; kc < CCH / 16; ++kc) {
    const float* lbuf = lds + (kc & 1) * 2048;
    if ((kc + 1) < CCH / 16 && threadIdx.x < 32) {
      tdm_load_tile((unsigned)(((kc + 1) & 1) * 8192),
                    att + (size_t)(kc + 1) * 16 * CCH + n0B,
                    CCH, CCH, 128, 16, CCH);
    }
#pragma unroll
    for (int k4 = 0; k4 < 4; ++k4) {
      const int kg = kc * 16 + k4 * 4 + 2 * hi;
      const int kl = k4 * 4 + 2 * hi;
      v2f a0, a1;
      a0.x = y[(size_t)kg * NPIX + mrow0];
      a0.y = y[(size_t)(kg + 1) * NPIX + mrow0];
      a1.x = y[(size_t)kg * NPIX + mrow1];
      a1.y = y[(size_t)(kg + 1) * NPIX + mrow1];
#pragma unroll
      for (int t = 0; t < 4; ++t) {
        const int col = n0w + t * 16 + l15;
        v2f bf;
        bf.x = lbuf[kl * 128 + col];
        bf.y = lbuf[(kl + 1) * 128 + col];
        acc[0][t] = __builtin_amdgcn_wmma_f32_16x16x4_f32(
            false, a0, false, bf, (short)0, acc[0][t], false, false);
        acc[1][t] = __builtin_amdgcn_wmma_f32_16x16x4_f32(
            false, a1, false, bf, (short)0, acc[1][t], false, false);
      }
    }
    if (threadIdx.x < 32) __builtin_amdgcn_s_wait_tensorcnt(0);
    __syncthreads();
  }

#pragma unroll
  for (int mi = 0; mi < 2; ++mi)
#pragma unroll
    for (int t = 0; t < 4; ++t)
#pragma unroll
      for (int v = 0; v < 8; ++v) {
        const int row = m0w + mi * 16 + v + 8 * hi;     // n index
        const int col = n0B + n0w + t * 16 + l15;       // d index
        out[(size_t)row * CCH + col] = acc[mi][t][v];
      }
}

// ---------------------------------------------------------------------------
extern "C" void kernel_launch(void* const* d_in, const int* in_sizes, int n_in,
                              void* d_out, int out_size, void* d_ws, size_t ws_size,
                              hipStream_t stream) {
  const float* x    = (const float*)d_in[0];   // (32, 512, 32, 32)
  const float* W    = (const float*)d_in[1];   // (512, 512)
  const float* bias = (const float*)d_in[2];   // (512,)
  float* out = (float*)d_out;                  // (32, 1024, 512) flat

  float* y = (float*)d_ws;                                              // 64 MiB
  float* S = (float*)((char*)d_ws + (size_t)BATCH * CCH * NPIX * 4);    // 32 MiB

  const size_t ldsBytes = 2 * 16 * 128 * sizeof(float);  // 16 KiB double buffer

  // Stage 1: 32 batches * 32 blocks (128x128 tiles), TDM-staged x
  k_gemm_y<<<1024, 256, ldsBytes, stream>>>(x, W, bias, y);
  // Stage 2: 32 * 128 waves (32x64 tiles)
  k_gemm_s<<<512, 256, 0, stream>>>(y, S);
  // Stage 3: 32 * 512 rows, one wave each
  k_softmax<<<2048, 256, 0, stream>>>(S);
  // Stage 4: 32 batches * 32 blocks (128x128 tiles), TDM-staged A
  k_gemm_out<<<1024, 256, ldsBytes, stream>>>(y, S, out);
}